// Model_34119220199556
// MI455X (gfx1250) — compile-verified
//
#include <hip/hip_runtime.h>
#include <math.h>

// CRF forward (log partition) for K=64 tags, N=500000 steps.
// Strategy: log-semiring scan  alpha_t = M_t (x) alpha_{t-1},  M_t = T + feat_t 1^T.
// 1) crf_chunk  : P=256 workgroups; each computes its chunk's 64x64 semiring
//    product in *linear* space with power-of-2 rescaling:
//        A_hat <- diag(exp(feat_t)) * (exp(T) * A_hat)
//    The 64x64x64 FP32 GEMM per step runs on V_WMMA_F32_16X16X4_F32
//    (256 wmma / step across 8 waves). A_hat lives in LDS in B-operand-native
//    interleaved layout phys(k,n) = ((k>>1)*64+n)*2 + (k&1) so each WMMA B
//    operand is one 8B-aligned ds_load_b64 (no repacking movs) and D writes
//    pair into ds_store_b64.
// 2) crf_combine: 8-level tree of log-semiring 64x64 matmuls (in-place in ws).
// 3) crf_finalize: apply init vector (START=62) and terminal row (STOP=63).

typedef float v2f __attribute__((ext_vector_type(2)));
typedef float v8f __attribute__((ext_vector_type(8)));

#define KTAGS 64
#define START_TAG 62
#define STOP_TAG 63
#define NCHUNKS 256

__device__ __forceinline__ int phys(int k, int n) {
  // interleave rows k, k+1 at consecutive floats for a given column n
  return ((k >> 1) * 64 + n) * 2 + (k & 1);
}

__global__ void __launch_bounds__(256) crf_chunk(const float* __restrict__ feats,
                                                 const float* __restrict__ trans,
                                                 float* __restrict__ G,
                                                 int N, int L0) {
  __shared__ float Ash[KTAGS * KTAGS];  // chunk-prefix matrix, interleaved layout
  __shared__ float fsh[KTAGS];          // exp(feat_t)
  __shared__ unsigned maxbits;
  __shared__ float logoff;

  const int tid  = threadIdx.x;
  const int lane = tid & 31;
  const int wave = tid >> 5;       // 0..7
  const int c16  = lane & 15;      // N-col within tile / M-row for A operand
  const int h    = lane >> 4;      // lane half
  const int mt   = wave >> 1;      // output row-tile 0..3
  const int n0   = ((wave & 1) * 2) * 16 + c16;  // first of two col-tiles
  const int n1   = n0 + 16;
  const int row  = mt * 16 + c16;  // E row held by this lane

  // Persistent A operands: E = exp(T), rows of tile mt, all K (16 chunks of 4).
  v2f Ereg[16];
#pragma unroll
  for (int kc = 0; kc < 16; ++kc) {
    Ereg[kc].x = __expf(trans[row * 64 + kc * 4 + 2 * h + 0]);
    Ereg[kc].y = __expf(trans[row * 64 + kc * 4 + 2 * h + 1]);
  }

  // A_hat = I (identity of the linear-space product)
#pragma unroll
  for (int q = 0; q < 16; ++q) {
    const int l = tid + 256 * q;           // logical (k,n)
    const int k = l >> 6, n = l & 63;
    Ash[phys(k, n)] = (k == n) ? 1.0f : 0.0f;
  }
  if (tid == 0) { logoff = 0.0f; maxbits = 0u; }
  __syncthreads();

  const int t0 = blockIdx.x * L0;
  int t1 = t0 + L0; if (t1 > N) t1 = N;

  for (int t = t0; t < t1; ++t) {
    if (tid < 64) fsh[tid] = __expf(feats[(size_t)t * 64 + tid]);

    // D = E * A_hat  (two 16x16 output tiles per wave, K=64 as 16 x k4)
    v8f acc0 = {}; v8f acc1 = {};
#pragma unroll
    for (int kc = 0; kc < 16; ++kc) {
      const int pb = (kc * 2 + h) * 128;   // phys row-pair base for k0 = kc*4+2h
      const v2f b0 = *reinterpret_cast<const v2f*>(&Ash[pb + 2 * n0]);
      const v2f b1 = *reinterpret_cast<const v2f*>(&Ash[pb + 2 * n1]);
      acc0 = __builtin_amdgcn_wmma_f32_16x16x4_f32(false, Ereg[kc], false, b0,
                                                   (short)0, acc0, false, false);
      acc1 = __builtin_amdgcn_wmma_f32_16x16x4_f32(false, Ereg[kc], false, b1,
                                                   (short)0, acc1, false, false);
    }
    __syncthreads();  // all B reads of old A_hat done; fsh visible

    // scale rows by exp(feat_t) and store the new A_hat (paired b64 stores)
    const int base = mt * 16 + 8 * h;      // even
    float f8[8];
#pragma unroll
    for (int r = 0; r < 8; ++r) f8[r] = fsh[base + r];
#pragma unroll
    for (int r2 = 0; r2 < 4; ++r2) {
      const int p = ((base >> 1) + r2) * 128;
      v2f s0; s0.x = acc0[2 * r2] * f8[2 * r2]; s0.y = acc0[2 * r2 + 1] * f8[2 * r2 + 1];
      v2f s1; s1.x = acc1[2 * r2] * f8[2 * r2]; s1.y = acc1[2 * r2 + 1] * f8[2 * r2 + 1];
      *reinterpret_cast<v2f*>(&Ash[p + 2 * n0]) = s0;
      *reinterpret_cast<v2f*>(&Ash[p + 2 * n1]) = s1;
    }

    // power-of-two renormalization every 8 steps (growth ~2^7 per step)
    if (((t - t0) & 7) == 7) {
      if (tid == 0) maxbits = 0u;
      __syncthreads();
      float m = 0.0f;
#pragma unroll
      for (int q = 0; q < 16; ++q) m = fmaxf(m, Ash[tid + 256 * q]);
      atomicMax(&maxbits, __float_as_uint(m));  // non-negative: bit order == value order
      __syncthreads();
      const float M = __uint_as_float(maxbits);
      int e = 0; (void)frexpf(M, &e);
      const float s = ldexpf(1.0f, -e);
#pragma unroll
      for (int q = 0; q < 16; ++q) Ash[tid + 256 * q] *= s;
      if (tid == 0) logoff += (float)e * 0.6931471805599453f;
    }
    __syncthreads();
  }

  // emit chunk matrix row-major in log space (+ accumulated offset)
  const float lo = logoff;
  float* __restrict__ Gc = G + (size_t)blockIdx.x * 4096;
#pragma unroll
  for (int q = 0; q < 16; ++q) {
    const int l = tid + 256 * q;           // logical row-major index
    const int k = l >> 6, n = l & 63;
    const float x = Ash[phys(k, n)];
    Gc[l] = (x > 0.0f) ? (__logf(x) + lo) : -1.0e30f;
  }
}

// slot[c] <- slot[c+stride] (x) slot[c]   (log-semiring matmul, later chunk left)
__global__ void __launch_bounds__(256) crf_combine(float* __restrict__ G, int stride) {
  __shared__ float A[4096];   // earlier chunk
  __shared__ float Bm[4096];  // later chunk
  const int tid = threadIdx.x;
  const size_t c = (size_t)blockIdx.x * (2 * stride);
  const float* __restrict__ Ag = G + c * 4096;
  const float* __restrict__ Bg = G + (c + stride) * 4096;
#pragma unroll
  for (int q = 0; q < 16; ++q) {
    A[tid + 256 * q]  = Ag[tid + 256 * q];
    Bm[tid + 256 * q] = Bg[tid + 256 * q];
  }
  __syncthreads();
  const int i  = tid >> 2;
  const int kb = (tid & 3) * 16;
  float* __restrict__ Cg = G + c * 4096;
  for (int o = 0; o < 16; ++o) {
    const int k = kb + o;
    float mm = -3.0e38f;
    for (int j = 0; j < 64; ++j) mm = fmaxf(mm, Bm[i * 64 + j] + A[j * 64 + k]);
    float s = 0.0f;
    for (int j = 0; j < 64; ++j) s += __expf(Bm[i * 64 + j] + A[j * 64 + k] - mm);
    Cg[i * 64 + k] = mm + __logf(s);
  }
}

__global__ void crf_finalize(const float* __restrict__ G,
                             const float* __restrict__ trans,
                             float* __restrict__ out) {
  __shared__ float red[64];
  const int i = threadIdx.x;  // 0..63
  float mm = -3.0e38f;
  for (int j = 0; j < 64; ++j) {
    const float init = (j == START_TAG) ? 0.0f : -10000.0f;
    mm = fmaxf(mm, G[i * 64 + j] + init);
  }
  float s = 0.0f;
  for (int j = 0; j < 64; ++j) {
    const float init = (j == START_TAG) ? 0.0f : -10000.0f;
    s += __expf(G[i * 64 + j] + init - mm);
  }
  red[i] = mm + __logf(s) + trans[STOP_TAG * 64 + i];
  __syncthreads();
  if (i == 0) {
    float M = -3.0e38f;
    for (int j = 0; j < 64; ++j) M = fmaxf(M, red[j]);
    float ss = 0.0f;
    for (int j = 0; j < 64; ++j) ss += __expf(red[j] - M);
    out[0] = M + __logf(ss);
  }
}

extern "C" void kernel_launch(void* const* d_in, const int* in_sizes, int n_in,
                              void* d_out, int out_size, void* d_ws, size_t ws_size,
                              hipStream_t stream) {
  (void)n_in; (void)out_size; (void)ws_size;
  const float* feats = (const float*)d_in[0];  // [N,64] f32
  const float* trans = (const float*)d_in[1];  // [64,64] f32
  float* G = (float*)d_ws;                     // NCHUNKS * 4096 floats = 4 MB

  const int N  = in_sizes[0] / KTAGS;          // 500000
  const int L0 = (N + NCHUNKS - 1) / NCHUNKS;  // steps per chunk

  crf_chunk<<<NCHUNKS, 256, 0, stream>>>(feats, trans, G, N, L0);
  for (int s = 1; s < NCHUNKS; s <<= 1)
    crf_combine<<<NCHUNKS / (2 * s), 256, 0, stream>>>(G, s);
  crf_finalize<<<1, 64, 0, stream>>>(G, trans, (float*)d_out);
}